// PredModel_13511967113603
// MI455X (gfx1250) — compile-verified
//
#include <hip/hip_runtime.h>
#include <cstdint>
#include <cstddef>

#define DFEAT 64
#define RREL  3

typedef __attribute__((ext_vector_type(2))) float v2f;
typedef __attribute__((ext_vector_type(8))) float v8f;

// ---------------------------------------------------------------------------
// Degree accumulation: deg[r*N + dst] += 1  (hardware f32 atomics)
// ---------------------------------------------------------------------------
__global__ void degree_kernel(const int* __restrict__ dst, float* __restrict__ deg,
                              int N, int E, int total) {
  int i = blockIdx.x * blockDim.x + threadIdx.x;
  if (i >= total) return;
  int r = i / E;
  unsafeAtomicAdd(&deg[(size_t)r * N + dst[i]], 1.0f);
}

// deg -> 1/max(deg,1), in place
__global__ void deginv_kernel(float* __restrict__ deg, int total) {
  int i = blockIdx.x * blockDim.x + threadIdx.x;
  if (i >= total) return;
  float d = deg[i];
  deg[i] = 1.0f / (d < 1.0f ? 1.0f : d);
}

// ---------------------------------------------------------------------------
// Scatter: m[r][dst] += h[src]   (16 lanes per edge, float4 per lane)
// ---------------------------------------------------------------------------
__global__ void scatter_kernel(const float* __restrict__ h,
                               const int* __restrict__ src,
                               const int* __restrict__ dst,
                               float* __restrict__ m,
                               int N, int E, int totalSlots) {
  int idx  = blockIdx.x * blockDim.x + threadIdx.x;
  int slot = idx >> 4;
  if (slot >= totalSlots) return;
  int lane = idx & 15;
  int r = slot / E;
  int s = src[slot];
  int d = dst[slot];
  const float4 v = ((const float4*)(h + (size_t)s * DFEAT))[lane];
  float* out = m + ((size_t)r * N + d) * DFEAT + lane * 4;
  unsafeAtomicAdd(out + 0, v.x);
  unsafeAtomicAdd(out + 1, v.y);
  unsafeAtomicAdd(out + 2, v.z);
  unsafeAtomicAdd(out + 3, v.w);
}

// ---------------------------------------------------------------------------
// hout = relu( sum_r (m[r] * deg_inv[r]) @ W[r] + b[r] )
// One wave computes a 16x64 output tile via V_WMMA_F32_16X16X4_F32.
// Block = 256 threads = 8 waves = 128 rows.
//
// W is staged into LDS K-pair-interleaved: sW[r][k/2][col][{k, k+1}] so each
// lane's B fragment {W[k][col], W[k+1][col]} is one aligned 64-bit LDS load
// (no register repacking before v_wmma).
//
// Fragment layouts (ISA 7.12.2, f32 16x16x4):
//   A (16x4, MxK): lane l<16 : rows M=l,    {K0,K1}; lane l>=16: {K2,K3}
//   B (4x16, KxN): lane l<16 : cols N=l,    {K0,K1}; lane l>=16: {K2,K3}
//   C/D (16x16):   vgpr v, lane l -> M = v + 8*(l>=16), N = l&15
// ---------------------------------------------------------------------------
__global__ __launch_bounds__(256)
void hetero_gemm_relu(const float* __restrict__ m,
                      const float* __restrict__ deg_inv,
                      const float* __restrict__ W,
                      const float* __restrict__ b,
                      float* __restrict__ hout, int N) {
  // [r][k>>1][col][k&1]
  __shared__ float sW[RREL][DFEAT / 2][DFEAT][2];
  __shared__ float sB[RREL][DFEAT];
  for (int i = threadIdx.x; i < RREL * DFEAT * DFEAT; i += 256) {
    int r   = i >> 12;          // /4096
    int k   = (i >> 6) & 63;    // row of W (input-feature index)
    int col = i & 63;           // output-feature index
    sW[r][k >> 1][col][k & 1] = W[i];
  }
  for (int i = threadIdx.x; i < RREL * DFEAT; i += 256)
    (&sB[0][0])[i] = b[i];
  __syncthreads();

  const int wave = threadIdx.x >> 5;
  const int lane = threadIdx.x & 31;
  const int l15  = lane & 15;
  const int hi   = lane >> 4;                 // 0: K0/K1 half, 1: K2/K3 half
  const int row0 = (blockIdx.x * 8 + wave) * 16;
  if (row0 >= N) return;                      // wave-uniform exit, EXEC stays full

  int aRow = row0 + l15;
  if (aRow > N - 1) aRow = N - 1;             // clamp reads; stores predicated below

  v8f acc[4] = {};                            // 4 column tiles of 16 -> 64 cols

  for (int r = 0; r < RREL; ++r) {
    const float dinv  = deg_inv[(size_t)r * N + aRow];
    const float* mrow = m + ((size_t)r * N + aRow) * DFEAT;
#pragma unroll
    for (int kk = 0; kk < DFEAT / 4; ++kk) {
      const int kb    = kk * 4 + hi * 2;      // lane's first K index
      const int kpair = kb >> 1;              // = kk*2 + hi
      v2f A;
      A.x = mrow[kb + 0] * dinv;
      A.y = mrow[kb + 1] * dinv;
#pragma unroll
      for (int nt = 0; nt < 4; ++nt) {
        const v2f B = *(const v2f*)&sW[r][kpair][nt * 16 + l15][0];
        acc[nt] = __builtin_amdgcn_wmma_f32_16x16x4_f32(
            false, A, false, B, (short)0, acc[nt], false, false);
      }
    }
  }

#pragma unroll
  for (int nt = 0; nt < 4; ++nt) {
    const int col   = nt * 16 + l15;
    const float bsum = sB[0][col] + sB[1][col] + sB[2][col];
#pragma unroll
    for (int v = 0; v < 8; ++v) {
      const int row = row0 + hi * 8 + v;
      if (row < N) {
        float val = acc[nt][v] + bsum;
        hout[(size_t)row * DFEAT + col] = val > 0.0f ? val : 0.0f;
      }
    }
  }
}

// ---------------------------------------------------------------------------
// Edge scoring: out[e] = dot(h[src[e]], h[dst[e]])  (16 lanes per edge)
// ---------------------------------------------------------------------------
__global__ void score_kernel(const float* __restrict__ h,
                             const int* __restrict__ es,
                             const int* __restrict__ ed,
                             float* __restrict__ out, int ne) {
  int idx  = blockIdx.x * blockDim.x + threadIdx.x;
  int slot = idx >> 4;
  if (slot >= ne) return;
  int lane = idx & 15;
  const float4 a = ((const float4*)(h + (size_t)es[slot] * DFEAT))[lane];
  const float4 c = ((const float4*)(h + (size_t)ed[slot] * DFEAT))[lane];
  float s = a.x * c.x + a.y * c.y + a.z * c.z + a.w * c.w;
#pragma unroll
  for (int off = 8; off >= 1; off >>= 1)
    s += __shfl_xor(s, off, 32);              // xor<=8 stays within each 16-lane group
  if (lane == 0) out[slot] = s;
}

// ---------------------------------------------------------------------------
extern "C" void kernel_launch(void* const* d_in, const int* in_sizes, int n_in,
                              void* d_out, int out_size, void* d_ws, size_t ws_size,
                              hipStream_t stream) {
  const float* x  = (const float*)d_in[0];
  const float* Wl[RREL] = { (const float*)d_in[1], (const float*)d_in[3], (const float*)d_in[5] };
  const float* bl[RREL] = { (const float*)d_in[2], (const float*)d_in[4], (const float*)d_in[6] };
  const int* src = (const int*)d_in[7];
  const int* dst = (const int*)d_in[8];
  const int* ps  = (const int*)d_in[9];
  const int* pd  = (const int*)d_in[10];
  const int* ns  = (const int*)d_in[11];
  const int* nd  = (const int*)d_in[12];

  const int N  = in_sizes[0] / DFEAT;
  const int E  = in_sizes[7] / RREL;
  const int EP = in_sizes[9];
  const int EN = in_sizes[11];

  float* ws  = (float*)d_ws;
  float* deg = ws;                                  // R*N         (deg -> deg_inv in place)
  float* mbuf = deg + (size_t)RREL * N;             // R*N*64
  float* hA  = mbuf + (size_t)RREL * N * DFEAT;     // N*64
  float* hB  = hA + (size_t)N * DFEAT;              // N*64

  // ---- in-degree normalization (shared by all layers) ----
  hipMemsetAsync(deg, 0, (size_t)RREL * N * sizeof(float), stream);
  {
    int total = RREL * E;
    degree_kernel<<<(total + 255) / 256, 256, 0, stream>>>(dst, deg, N, E, total);
    int tn = RREL * N;
    deginv_kernel<<<(tn + 255) / 256, 256, 0, stream>>>(deg, tn);
  }

  // ---- 3 hetero-conv layers ----
  const float* hin = x;
  float* hout = hA;
  for (int l = 0; l < RREL; ++l) {
    hipMemsetAsync(mbuf, 0, (size_t)RREL * N * DFEAT * sizeof(float), stream);
    int slots = RREL * E;
    scatter_kernel<<<((slots * 16) + 255) / 256, 256, 0, stream>>>(
        hin, src, dst, mbuf, N, E, slots);
    hetero_gemm_relu<<<(N + 127) / 128, 256, 0, stream>>>(
        mbuf, deg, Wl[l], bl[l], hout, N);
    hin  = hout;
    hout = (hout == hA) ? hB : hA;
  }

  // ---- edge scoring ----
  float* out = (float*)d_out;
  score_kernel<<<((EP * 16) + 255) / 256, 256, 0, stream>>>(hin, ps, pd, out, EP);
  score_kernel<<<((EN * 16) + 255) / 256, 256, 0, stream>>>(hin, ns, nd, out + EP, EN);
}